// QLIFSpike_53549652246725
// MI455X (gfx1250) — compile-verified
//
#include <hip/hip_runtime.h>

// QLIF spike recurrence, MI455X (gfx1250, wave32).
// x: [B=32, T=8, C=128, H=32, W=32] fp32 -> spikes same shape.
// Memory-bound streaming kernel: one thread owns 4 consecutive W-elements and
// walks T serially. All 8 timestep loads issued up front (B128, non-temporal),
// stores are B128 non-temporal. ~268 MB total traffic -> ~11.5 us roofline.

typedef float v4f __attribute__((ext_vector_type(4)));   // native vector: OK for NT builtins

#define QLIF_T       8
#define QLIF_CHW4    32768          // (128*32*32)/4 float4 per (b,t) slice
#define QLIF_THRESH  1.0f
#define QLIF_TAU     0.5f
#define QLIF_NEG_LB  30.0f          // -LB
#define QLIF_MAXIDX  31.0f          // N_LEVELS - 1

__device__ __forceinline__ float qlif_step(float& mem, float xt) {
    // mem = mem*tau + x
    float m = fmaf(mem, QLIF_TAU, xt);
    // spike = (m - thresh) > 0
    float spike = (m > QLIF_THRESH) ? 1.0f : 0.0f;
    // reset on spike
    m = (spike > 0.0f) ? 0.0f : m;
    // quantize to {LB + i : i in [0,31]}; jnp.round == round-to-nearest-even == rintf
    float idx = fminf(fmaxf(rintf(m + QLIF_NEG_LB), 0.0f), QLIF_MAXIDX);
    mem = idx - QLIF_NEG_LB;
    return spike;
}

__global__ __launch_bounds__(256) void qlif_spike_kernel(
        const v4f* __restrict__ x, v4f* __restrict__ out) {
    const unsigned gid = blockIdx.x * blockDim.x + threadIdx.x;   // float4 site id
    const unsigned b   = gid >> 15;                 // / CHW4
    const unsigned r   = gid & (QLIF_CHW4 - 1);     // % CHW4
    const size_t base  = (size_t)b * (QLIF_T * QLIF_CHW4) + r;

    // Issue all 8 timestep loads up front: 8 outstanding B128 NT loads/thread.
    v4f xv[QLIF_T];
#pragma unroll
    for (int t = 0; t < QLIF_T; ++t) {
        xv[t] = __builtin_nontemporal_load(&x[base + (size_t)t * QLIF_CHW4]);
    }

    float mx = 0.0f, my = 0.0f, mz = 0.0f, mw = 0.0f;   // membrane state
#pragma unroll
    for (int t = 0; t < QLIF_T; ++t) {
        v4f s;
        s.x = qlif_step(mx, xv[t].x);
        s.y = qlif_step(my, xv[t].y);
        s.z = qlif_step(mz, xv[t].z);
        s.w = qlif_step(mw, xv[t].w);
        __builtin_nontemporal_store(s, &out[base + (size_t)t * QLIF_CHW4]);
    }
}

extern "C" void kernel_launch(void* const* d_in, const int* in_sizes, int n_in,
                              void* d_out, int out_size, void* d_ws, size_t ws_size,
                              hipStream_t stream) {
    (void)n_in; (void)out_size; (void)d_ws; (void)ws_size;
    const v4f* x = (const v4f*)d_in[0];
    v4f*       o = (v4f*)d_out;

    // total elements = B*T*C*H*W; each thread handles T steps * 4 elements.
    const int total    = in_sizes[0];                 // 33,554,432
    const int nthreads = total / (QLIF_T * 4);        // 1,048,576
    const int block    = 256;                         // 8 wave32 waves / WG
    const int grid     = (nthreads + block - 1) / block;   // 4096

    qlif_spike_kernel<<<grid, block, 0, stream>>>(x, o);
}